// LoongSpikeKernel_17351667876172
// MI455X (gfx1250) — compile-verified
//
#include <hip/hip_runtime.h>
#include <math.h>

typedef float v2f __attribute__((ext_vector_type(2)));
typedef float v8f __attribute__((ext_vector_type(8)));

#define HCH     512      // d_model
#define NSTATE  32       // complex states per mixture component
#define NP      64       // M * NST poles
#define LLEN    2048
#define LI      16       // coarse time index  (l = 128*i + j)
#define LJ      128      // fine time index
#define WSTRIDE 144      // padded row stride for W in LDS (bank-conflict free)

// One block per channel h. 256 threads = 8 wave32.
__global__ __launch_bounds__(256) void ssm_vandermonde_wmma(
    const float* __restrict__ C_real,      // [1, H, NST, 2]
    const float* __restrict__ log_dt,      // [H]
    const float* __restrict__ log_A_real,  // [H, NST]
    const float* __restrict__ A_imag,      // [H, NST]
    const float* __restrict__ omega_logit, // [2]
    const float* __restrict__ eta_logit,   // [2]
    float* __restrict__ out)               // [H, L]
{
    __shared__ float s_dar[NP], s_dai[NP], s_cdr[NP], s_cdi[NP];
    __shared__ float sUr [NP * LI];        // 2*Re(C_disc * exp(dtA*128*i))   [n][i]
    __shared__ float sUmi[NP * LI];        // -2*Im(...)                      [n][i]
    __shared__ float sWr [NP * WSTRIDE];   // Re(exp(dtA*j))                  [n][j]
    __shared__ float sWi [NP * WSTRIDE];   // Im(exp(dtA*j))                  [n][j]

    const int h   = blockIdx.x;
    const int tid = threadIdx.x;

    // ---- Phase 1: per-pole parameters (64 lanes) -------------------------
    if (tid < NP) {
        const int n = tid, m = n >> 5, k = n & 31;
        const float om = 1e-6f + (100.0f - 1e-6f) / (1.0f + expf(-omega_logit[m]));
        const float et = 1e-6f + (10.0f  - 1e-6f) / (1.0f + expf(-eta_logit[m]));
        // A_frac = -omega + eta * (-exp(logA) + i*Aimag)
        const float ar = -om - et * expf(log_A_real[h * NSTATE + k]);
        const float ai =        et * A_imag   [h * NSTATE + k];
        const float dt = expf(log_dt[h]);
        const float dar = ar * dt, dai = ai * dt;
        // C_frac = eta * C
        const float cfr = et * C_real[(h * NSTATE + k) * 2 + 0];
        const float cfi = et * C_real[(h * NSTATE + k) * 2 + 1];
        // C_disc = C_frac * (exp(dtA) - 1) / (A_frac + 1e-8), or C_frac*dt if |A_frac| small
        float cdr, cdi;
        if (sqrtf(ar * ar + ai * ai) < 1e-6f) {
            cdr = cfr * dt;  cdi = cfi * dt;
        } else {
            const float e = expf(dar);
            float sn, cs;  sincosf(dai, &sn, &cs);
            const float numr = e * cs - 1.0f, numi = e * sn;
            const float dr = ar + 1e-8f, di = ai;
            const float inv = 1.0f / (dr * dr + di * di);
            const float rr = (numr * dr + numi * di) * inv;   // num / den
            const float ri = (numi * dr - numr * di) * inv;
            cdr = cfr * rr - cfi * ri;
            cdi = cfr * ri + cfi * rr;
        }
        s_dar[n] = dar;  s_dai[n] = dai;  s_cdr[n] = cdr;  s_cdi[n] = cdi;
    }
    __syncthreads();

    // ---- Phase 2: build U (64x16) and W (64x128) factors -----------------
    for (int idx = tid; idx < NP * LI; idx += 256) {
        const int n = idx >> 4, i = idx & 15;
        const float t = 128.0f * (float)i;
        const float e = expf(s_dar[n] * t);
        float sn, cs;  sincosf(s_dai[n] * t, &sn, &cs);
        const float Er = e * cs, Ei = e * sn;
        sUr [n * LI + i] =  2.0f * (s_cdr[n] * Er - s_cdi[n] * Ei);
        sUmi[n * LI + i] = -2.0f * (s_cdr[n] * Ei + s_cdi[n] * Er);
    }
    for (int idx = tid; idx < NP * LJ; idx += 256) {
        const int n = idx >> 7, j = idx & 127;
        const float t = (float)j;
        const float e = expf(s_dar[n] * t);
        float sn, cs;  sincosf(s_dai[n] * t, &sn, &cs);
        sWr[n * WSTRIDE + j] = e * cs;
        sWi[n * WSTRIDE + j] = e * sn;
    }
    __syncthreads();

    // ---- Phase 3: K[i, j0+jj] = (2Ur)^T Wr + (-2Ui)^T Wi via f32 WMMA ----
    // V_WMMA_F32_16X16X4_F32 layouts (wave32):
    //   A 16x4 : lane half hh=lane>>4, M=lane&15, K = 2*hh + v  (v = frag elem)
    //   B 4x16 : N=lane&15,            K = 2*hh + v
    //   C/D    : VGPR r -> row r + 8*hh, col lane&15
    const int lane  = tid & 31;
    const int wv    = tid >> 5;          // 8 waves -> 8 column slabs of 16
    const int laneM = lane & 15;
    const int hh    = lane >> 4;
    const int j0    = wv * 16;

    v8f acc = {};
#pragma unroll
    for (int s = 0; s < 16; ++s) {
        const int nb = 4 * s + 2 * hh;   // K-base for this lane half
        v2f a, b;
        a.x = sUr[(nb + 0) * LI + laneM];
        a.y = sUr[(nb + 1) * LI + laneM];
        b.x = sWr[(nb + 0) * WSTRIDE + j0 + laneM];
        b.y = sWr[(nb + 1) * WSTRIDE + j0 + laneM];
        acc = __builtin_amdgcn_wmma_f32_16x16x4_f32(false, a, false, b,
                                                    (short)0, acc, false, false);
        a.x = sUmi[(nb + 0) * LI + laneM];
        a.y = sUmi[(nb + 1) * LI + laneM];
        b.x = sWi[(nb + 0) * WSTRIDE + j0 + laneM];
        b.y = sWi[(nb + 1) * WSTRIDE + j0 + laneM];
        acc = __builtin_amdgcn_wmma_f32_16x16x4_f32(false, a, false, b,
                                                    (short)0, acc, false, false);
    }

#pragma unroll
    for (int r = 0; r < 8; ++r) {
        const int i = r + 8 * hh;                    // coarse index
        const int l = 128 * i + j0 + laneM;          // l = 128*i + j
        out[h * LLEN + l] = acc[r];
    }
}

extern "C" void kernel_launch(void* const* d_in, const int* in_sizes, int n_in,
                              void* d_out, int out_size, void* d_ws, size_t ws_size,
                              hipStream_t stream) {
    (void)in_sizes; (void)n_in; (void)out_size; (void)d_ws; (void)ws_size;
    const float* C_real      = (const float*)d_in[0];
    const float* log_dt      = (const float*)d_in[1];
    const float* log_A_real  = (const float*)d_in[2];
    const float* A_imag      = (const float*)d_in[3];
    const float* omega_logit = (const float*)d_in[4];
    const float* eta_logit   = (const float*)d_in[5];
    // d_in[6] = L (defaults to 2048; tiling is specialized for L_DEFAULT)
    float* out = (float*)d_out;

    ssm_vandermonde_wmma<<<HCH, 256, 0, stream>>>(
        C_real, log_dt, log_A_real, A_imag, omega_logit, eta_logit, out);
}